// OverlapGAT_1597727834335
// MI455X (gfx1250) — compile-verified
//
#include <hip/hip_runtime.h>
#include <math.h>

typedef __attribute__((ext_vector_type(2))) float v2f;
typedef __attribute__((ext_vector_type(8))) float v8f;

// ---------------------------------------------------------------------------
// Generic direct convolution (kernel W-dim = 1) + ReLU.  Layout NCHW.
// ---------------------------------------------------------------------------
__global__ __launch_bounds__(256) void conv2d_relu(
    const float* __restrict__ in, const float* __restrict__ w,
    float* __restrict__ out,
    int B, int IC, int IH, int OC, int OH, int W, int KH, int SH)
{
    long idx = (long)blockIdx.x * blockDim.x + threadIdx.x;
    long total = (long)B * OC * OH * W;
    if (idx >= total) return;
    int x = (int)(idx % W); long r = idx / W;
    int oh = (int)(r % OH); r /= OH;
    int oc = (int)(r % OC); int b = (int)(r / OC);
    const float* wp = w + (long)oc * IC * KH;
    const float* ip = in + ((long)(b * IC) * IH + (long)oh * SH) * W + x;
    float acc = 0.f;
    for (int ic = 0; ic < IC; ++ic)
        for (int kh = 0; kh < KH; ++kh)
            acc += ip[((long)ic * IH + kh) * W] * wp[ic * KH + kh];
    out[idx] = fmaxf(acc, 0.f);
}

// ---------------------------------------------------------------------------
// Fragment loader: two consecutive-K elements at stride s.
// CONTIG => s == 1 and 8B-aligned: single b64 vector load.
// ---------------------------------------------------------------------------
template <bool CONTIG>
__device__ __forceinline__ v2f ld2(const float* __restrict__ p, long s)
{
    if constexpr (CONTIG) {
        return *(const v2f*)p;
    } else {
        v2f r; r.x = p[0]; r.y = p[s]; return r;
    }
}

// ---------------------------------------------------------------------------
// Generic strided fp32 WMMA GEMM:  C[m,n] = act( sum_k A[m,k]*B[k,n] + bias[n] )
// A[m*Ars + k*Acs], B[k*Brs + n*Bcs], C[m*Crs + n*Ccs]; batch via blockIdx.z.
// Block = 256 threads = 8 waves arranged 4(M)x2(N); each wave owns a 16x32
// C tile (two accumulators sharing one A fragment) -> 64x64 block tile.
// K-loop: branch-free (addresses clamped, invalid lanes' results never stored;
// EXEC stays all-ones through the WMMAs), pointer-increment addressing,
// unrolled x2 -> 4 v_wmma per iteration. Requires K % 4 == 0.
// AC/BC: Acs==1 / Brs==1 known at compile time -> b64 fragment loads.
// ---------------------------------------------------------------------------
template <bool AC, bool BC>
__global__ __launch_bounds__(256) void gemm_wmma_f32(
    const float* __restrict__ A, const float* __restrict__ Bm,
    float* __restrict__ C,
    int M, int N, int K,
    long Ars, long Acs, long Brs, long Bcs, long Crs, long Ccs,
    long Abat, long Bbat, long Cbat,
    const float* __restrict__ bias, int relu)
{
    int tid = threadIdx.x;
    int wave = tid >> 5;
    int lane = tid & 31;
    int wrow = wave >> 1;      // 0..3
    int wcol = wave & 1;       // 0..1
    int m0 = (blockIdx.y * 4 + wrow) * 16;
    int n0 = blockIdx.x * 64 + wcol * 32;
    const float* Ab = A + (long)blockIdx.z * Abat;
    const float* Bb = Bm + (long)blockIdx.z * Bbat;
    float*       Cb = C + (long)blockIdx.z * Cbat;

    const long acs = AC ? 1 : Acs;
    const long brs = BC ? 1 : Brs;

    int mi  = lane & 15;
    int khi = lane >> 4;       // 0/1 -> covers K offsets {0,1} / {2,3}
    int row  = m0 + mi;
    int col0 = n0 + mi;
    int col1 = n0 + 16 + mi;
    // clamp addresses for memory safety; out-of-range lanes feed only
    // out-of-range C rows/cols, whose stores are guarded off below.
    long rowC  = (row  < M) ? row  : 0;
    long col0C = (col0 < N) ? col0 : 0;
    long col1C = (col1 < N) ? col1 : 0;

    const float* pa  = Ab + rowC  * Ars + (long)(2 * khi) * acs;
    const float* pb0 = Bb + col0C * Bcs + (long)(2 * khi) * brs;
    const float* pb1 = Bb + col1C * Bcs + (long)(2 * khi) * brs;
    const long stepA = 4 * acs;
    const long stepB = 4 * brs;

    v8f acc0 = {};
    v8f acc1 = {};
    int k0 = 0;
    for (; k0 + 8 <= K; k0 += 8) {
        v2f a0  = ld2<AC>(pa, acs);
        v2f b00 = ld2<BC>(pb0, brs);
        v2f b10 = ld2<BC>(pb1, brs);
        v2f a1  = ld2<AC>(pa + stepA, acs);
        v2f b01 = ld2<BC>(pb0 + stepB, brs);
        v2f b11 = ld2<BC>(pb1 + stepB, brs);
        pa += 2 * stepA; pb0 += 2 * stepB; pb1 += 2 * stepB;
        acc0 = __builtin_amdgcn_wmma_f32_16x16x4_f32(false, a0, false, b00, (short)0, acc0, false, false);
        acc1 = __builtin_amdgcn_wmma_f32_16x16x4_f32(false, a0, false, b10, (short)0, acc1, false, false);
        acc0 = __builtin_amdgcn_wmma_f32_16x16x4_f32(false, a1, false, b01, (short)0, acc0, false, false);
        acc1 = __builtin_amdgcn_wmma_f32_16x16x4_f32(false, a1, false, b11, (short)0, acc1, false, false);
    }
    for (; k0 < K; k0 += 4) {
        v2f a  = ld2<AC>(pa, acs);
        v2f b0 = ld2<BC>(pb0, brs);
        v2f b1 = ld2<BC>(pb1, brs);
        pa += stepA; pb0 += stepB; pb1 += stepB;
        acc0 = __builtin_amdgcn_wmma_f32_16x16x4_f32(false, a, false, b0, (short)0, acc0, false, false);
        acc1 = __builtin_amdgcn_wmma_f32_16x16x4_f32(false, a, false, b1, (short)0, acc1, false, false);
    }

    float bv0 = (bias && col0 < N) ? bias[col0] : 0.f;
    float bv1 = (bias && col1 < N) ? bias[col1] : 0.f;
#pragma unroll
    for (int j = 0; j < 8; ++j) {
        int r = m0 + j + 8 * khi;
        if (r < M) {
            if (col0 < N) {
                float v = acc0[j] + bv0;
                if (relu) v = fmaxf(v, 0.f);
                Cb[(long)r * Crs + (long)col0 * Ccs] = v;
            }
            if (col1 < N) {
                float v = acc1[j] + bv1;
                if (relu) v = fmaxf(v, 0.f);
                Cb[(long)r * Crs + (long)col1 * Ccs] = v;
            }
        }
    }
}

// ---------------------------------------------------------------------------
// GAT attention scores: as/ad[bn,h] = sum_d xh[bn, h*256+d] * att[h,d]
// grid = B*N blocks; 8 waves per block = (4 heads) x (src,dst)
// ---------------------------------------------------------------------------
__global__ __launch_bounds__(256) void att_scores(
    const float* __restrict__ xh, const float* __restrict__ att_src,
    const float* __restrict__ att_dst,
    float* __restrict__ as_, float* __restrict__ ad_)
{
    int bn = blockIdx.x;
    int wave = threadIdx.x >> 5;
    int lane = threadIdx.x & 31;
    int h = wave >> 1;
    int which = wave & 1;
    const float* att = which ? att_dst : att_src;
    const float* rowp = xh + (long)bn * 1024 + h * 256;
    float s = 0.f;
    for (int d = lane; d < 256; d += 32) s += rowp[d] * att[h * 256 + d];
    for (int off = 16; off > 0; off >>= 1) s += __shfl_down(s, off, 32);
    if (lane == 0) (which ? ad_ : as_)[bn * 4 + h] = s;
}

// ---------------------------------------------------------------------------
// Deterministic CSR build (no atomics): each dst thread scans the edge list.
// edge_index flat: [0..E) = src, [E..2E) = dst
// ---------------------------------------------------------------------------
__global__ void csr_count(const int* __restrict__ ei, int E, int* __restrict__ deg, int N)
{
    int d = blockIdx.x * blockDim.x + threadIdx.x;
    if (d >= N) return;
    const int* dsts = ei + E;
    int c = 0;
    for (int e = 0; e < E; ++e) c += (dsts[e] == d);
    deg[d] = c;
}
__global__ void csr_scan(const int* __restrict__ deg, int* __restrict__ off, int N)
{
    if (blockIdx.x == 0 && threadIdx.x == 0) {
        int s = 0;
        for (int i = 0; i < N; ++i) { off[i] = s; s += deg[i]; }
        off[N] = s;
    }
}
__global__ void csr_fill(const int* __restrict__ ei, int E,
                         const int* __restrict__ off, int* __restrict__ csrc, int N)
{
    int d = blockIdx.x * blockDim.x + threadIdx.x;
    if (d >= N) return;
    const int* srcs = ei;
    const int* dsts = ei + E;
    int p = off[d];
    for (int e = 0; e < E; ++e)
        if (dsts[e] == d) csrc[p++] = srcs[e];
}

// ---------------------------------------------------------------------------
// GAT segment softmax + aggregation + head-mean + bias.
// Block per (b,dst); 256 threads (one per output dim). Writes the second half
// of the 512-wide concat buffer.
// ---------------------------------------------------------------------------
__global__ __launch_bounds__(256) void gat_aggregate(
    const float* __restrict__ xh, const float* __restrict__ as_,
    const float* __restrict__ ad_,
    const int* __restrict__ off, const int* __restrict__ csrc,
    const float* __restrict__ bias, float* __restrict__ cat, int Nn)
{
    int d = blockIdx.x;
    int b = blockIdx.y;
    __shared__ int   sSrc[128];
    __shared__ float sAl[4 * 128];
    int o0 = off[d];
    int deg = off[d + 1] - o0;
    if (deg > 128) deg = 128;
    int t = threadIdx.x;
    if (t < deg) sSrc[t] = csrc[o0 + t];
    __syncthreads();
    for (int idx = t; idx < 4 * deg; idx += 256) {
        int h = idx / deg, j = idx - h * deg;
        float e = as_[(b * Nn + sSrc[j]) * 4 + h] + ad_[(b * Nn + d) * 4 + h];
        e = (e > 0.f) ? e : 0.2f * e;                 // leaky_relu(0.2)
        sAl[h * 128 + j] = e;
    }
    __syncthreads();
    if (t < 4) {
        int h = t;
        float m = -INFINITY;
        for (int j = 0; j < deg; ++j) m = fmaxf(m, sAl[h * 128 + j]);
        float s = 0.f;
        for (int j = 0; j < deg; ++j) s += expf(sAl[h * 128 + j] - m);
        float inv = 1.f / (s + 1e-16f);
        for (int j = 0; j < deg; ++j)
            sAl[h * 128 + j] = expf(sAl[h * 128 + j] - m) * inv;
    }
    __syncthreads();
    float acc = bias[t];
    const float* xb = xh + (long)(b * Nn) * 1024 + t;
    for (int h = 0; h < 4; ++h) {
        float p = 0.f;
        const float* base = xb + h * 256;
        for (int j = 0; j < deg; ++j)
            p += sAl[h * 128 + j] * base[(long)sSrc[j] * 1024];
        acc += 0.25f * p;                              // head mean
    }
    cat[((long)(b * Nn + d)) * 512 + 256 + t] = acc;
}

// ---------------------------------------------------------------------------
// L2 normalize contiguous rows in place (row = blockIdx.x, length rowlen).
// ---------------------------------------------------------------------------
__global__ __launch_bounds__(256) void l2norm_rows(float* __restrict__ x, int rowlen)
{
    __shared__ float red[256];
    long base = (long)blockIdx.x * rowlen;
    int t = threadIdx.x;
    float s = 0.f;
    for (int i = t; i < rowlen; i += 256) { float v = x[base + i]; s += v * v; }
    red[t] = s; __syncthreads();
    for (int o = 128; o > 0; o >>= 1) { if (t < o) red[t] += red[t + o]; __syncthreads(); }
    float inv = 1.f / fmaxf(sqrtf(red[0]), 1e-12f);
    for (int i = t; i < rowlen; i += 256) x[base + i] *= inv;
}

// L2 normalize vlad columns: block per (b,k); elements vlad[b*65536 + f*64 + k]
__global__ __launch_bounds__(256) void l2norm_cols(float* __restrict__ vlad)
{
    __shared__ float red[256];
    int b = blockIdx.x >> 6, k = blockIdx.x & 63;
    float* base = vlad + (long)b * 65536 + k;
    int t = threadIdx.x;
    float s = 0.f;
    for (int f = t; f < 1024; f += 256) { float v = base[(long)f * 64]; s += v * v; }
    red[t] = s; __syncthreads();
    for (int o = 128; o > 0; o >>= 1) { if (t < o) red[t] += red[t + o]; __syncthreads(); }
    float inv = 1.f / fmaxf(sqrtf(red[0]), 1e-12f);
    for (int f = t; f < 1024; f += 256) base[(long)f * 64] *= inv;
}

// softmax over K=64 in place; block(64) per (b,n)
__global__ __launch_bounds__(64) void softmax64(float* __restrict__ act)
{
    __shared__ float red[64];
    int t = threadIdx.x;
    long base = (long)blockIdx.x * 64;
    float v = act[base + t];
    red[t] = v; __syncthreads();
    for (int o = 32; o > 0; o >>= 1) { if (t < o) red[t] = fmaxf(red[t], red[t + o]); __syncthreads(); }
    float m = red[0]; __syncthreads();
    float e = expf(v - m);
    red[t] = e; __syncthreads();
    for (int o = 32; o > 0; o >>= 1) { if (t < o) red[t] += red[t + o]; __syncthreads(); }
    act[base + t] = e / red[0];
}

// asum[b,k] = sum_n act[b,n,k]; block(64) per b
__global__ __launch_bounds__(64) void act_colsum(const float* __restrict__ act,
                                                 float* __restrict__ asum, int Nn)
{
    int b = blockIdx.x, k = threadIdx.x;
    float s = 0.f;
    for (int n = 0; n < Nn; ++n) s += act[((long)(b * Nn + n)) * 64 + k];
    asum[b * 64 + k] = s;
}

// vlad[b,f,k] -= asum[b,k] * cw2[f,k]
__global__ __launch_bounds__(256) void vlad_sub(float* __restrict__ vlad,
                                                const float* __restrict__ asum,
                                                const float* __restrict__ cw2)
{
    int idx = blockIdx.x * blockDim.x + threadIdx.x;   // 4*65536 total
    int b = idx >> 16;
    int fk = idx & 65535;
    vlad[idx] -= asum[b * 64 + (fk & 63)] * cw2[fk];
}

// Deterministic split-K FC: partials[chunk][b][256]. 32 chunks of 2048.
__global__ __launch_bounds__(256) void hidden_partial(
    const float* __restrict__ vlad, const float* __restrict__ w,
    float* __restrict__ part)
{
    int t = threadIdx.x;
    int o = blockIdx.x * 64 + (t & 63);
    int chunk = blockIdx.y * 4 + (t >> 6);
    int i0 = chunk * 2048;
    float a0 = 0.f, a1 = 0.f, a2 = 0.f, a3 = 0.f;
    for (int i = i0; i < i0 + 2048; ++i) {
        float wv = w[(long)i * 256 + o];
        a0 += vlad[i]          * wv;
        a1 += vlad[65536 + i]  * wv;
        a2 += vlad[131072 + i] * wv;
        a3 += vlad[196608 + i] * wv;
    }
    float* p = part + (long)chunk * 1024 + o;
    p[0] = a0; p[256] = a1; p[512] = a2; p[768] = a3;
}
__global__ __launch_bounds__(256) void hidden_reduce(const float* __restrict__ part,
                                                     float* __restrict__ h1)
{
    int idx = blockIdx.x * blockDim.x + threadIdx.x;   // 1024
    if (idx >= 1024) return;
    float s = 0.f;
    for (int c = 0; c < 32; ++c) s += part[(long)c * 1024 + idx];
    h1[idx] = s;
}

// BatchNorm(train, B=4) + gating GEMM + sigmoid + final L2 norm. One block.
__global__ __launch_bounds__(256) void bn_gate_final(
    const float* __restrict__ h1, const float* __restrict__ gamma,
    const float* __restrict__ beta, const float* __restrict__ gw,
    const float* __restrict__ gb, float* __restrict__ out)
{
    __shared__ float vbn[1024];
    __shared__ float fo[1024];
    __shared__ float nrm[4];
    int o = threadIdx.x;
    float x0 = h1[o], x1 = h1[256 + o], x2 = h1[512 + o], x3 = h1[768 + o];
    float mu = (x0 + x1 + x2 + x3) * 0.25f;
    float d0 = x0 - mu, d1 = x1 - mu, d2 = x2 - mu, d3 = x3 - mu;
    float var = (d0 * d0 + d1 * d1 + d2 * d2 + d3 * d3) * 0.25f;
    float is = rsqrtf(var + 1e-5f) * gamma[o];
    vbn[o] = d0 * is + beta[o];
    vbn[256 + o] = d1 * is + beta[o];
    vbn[512 + o] = d2 * is + beta[o];
    vbn[768 + o] = d3 * is + beta[o];
    __syncthreads();
    for (int b = 0; b < 4; ++b) {
        float g = gb[o];
        for (int p = 0; p < 256; ++p) g += vbn[b * 256 + p] * gw[p * 256 + o];
        g = 1.f / (1.f + expf(-g));
        fo[b * 256 + o] = vbn[b * 256 + o] * g;
    }
    __syncthreads();
    if (o < 4) {
        float s = 0.f;
        for (int p = 0; p < 256; ++p) s += fo[o * 256 + p] * fo[o * 256 + p];
        nrm[o] = 1.f / fmaxf(sqrtf(s), 1e-12f);
    }
    __syncthreads();
    for (int b = 0; b < 4; ++b) out[b * 256 + o] = fo[b * 256 + o] * nrm[b];
}

// ---------------------------------------------------------------------------
// Workspace layout (float offsets)
// ---------------------------------------------------------------------------
static const long OFF_PING = 0;          // 3,456,000
static const long OFF_PONG = 3456000;    // 3,456,000
static const long OFF_CAT  = 6912000;    // 4*900*512
static const long OFF_XH   = 8755200;    // 4*900*1024
static const long OFF_AS   = 12441600;   // 4*900*4
static const long OFF_AD   = 12456000;   // 4*900*4
static const long OFF_OUTC = 12470400;   // 4*900*1024
static const long OFF_ACT  = 16156800;   // 4*900*64
static const long OFF_ASUM = 16387200;   // 256
static const long OFF_VLAD = 16387456;   // 4*65536
static const long OFF_PART = 16649600;   // 32*1024
static const long OFF_H1   = 16682368;   // 1024
static const long OFF_INT  = 16683392;   // int region starts here

extern "C" void kernel_launch(void* const* d_in, const int* in_sizes, int n_in,
                              void* d_out, int out_size, void* d_ws, size_t ws_size,
                              hipStream_t stream) {
    const float* x_l = (const float*)d_in[0];
    const float* convw[11];
    for (int i = 0; i < 11; ++i) convw[i] = (const float*)d_in[1 + i];
    const float* w_last1   = (const float*)d_in[12];   // (256,128)
    const float* w_last2   = (const float*)d_in[13];   // (1024,512)
    const float* gat_w     = (const float*)d_in[14];   // (256,4,256)
    const float* gat_asrc  = (const float*)d_in[15];   // (4,256)
    const float* gat_adst  = (const float*)d_in[16];
    const float* gat_bias  = (const float*)d_in[17];   // (256,)
    const float* cw        = (const float*)d_in[18];   // (1024,64)
    const float* cb        = (const float*)d_in[19];   // (64,)
    const float* cw2       = (const float*)d_in[20];   // (1,1024,64)
    const float* hw        = (const float*)d_in[21];   // (65536,256)
    const float* bn_g      = (const float*)d_in[22];
    const float* bn_b      = (const float*)d_in[23];
    const float* gate_w    = (const float*)d_in[24];   // (256,256)
    const float* gate_b    = (const float*)d_in[25];
    const int*   ei        = (const int*)d_in[26];     // [2,E]
    const int E = in_sizes[26] / 2;
    const int B = 4, Nn = 900;

    float* ws = (float*)d_ws;
    int*  iws = (int*)(ws + OFF_INT);
    int* deg  = iws;          // 900
    int* off  = iws + 1024;   // 901
    int* csrc = iws + 2048;   // E

    // ---- conv stack (ping-pong) ----
    struct CS { int oc, ic, kh, sh, ih, oh; };
    const CS cs[11] = {
        {16, 5, 5, 1, 64, 60}, {32, 16, 3, 2, 60, 29}, {64, 32, 3, 2, 29, 14},
        {64, 64, 3, 2, 14, 6}, {128, 64, 2, 2, 6, 3},  {128, 128, 1, 2, 3, 2},
        {128, 128, 1, 2, 2, 1},{128, 128, 1, 2, 1, 1}, {128, 128, 1, 2, 1, 1},
        {128, 128, 1, 2, 1, 1},{128, 128, 1, 2, 1, 1}};
    const float* cin = x_l;
    float* cout = nullptr;
    for (int i = 0; i < 11; ++i) {
        cout = ws + ((i & 1) ? OFF_PONG : OFF_PING);
        long total = (long)B * cs[i].oc * cs[i].oh * 900;
        conv2d_relu<<<(int)((total + 255) / 256), 256, 0, stream>>>(
            cin, convw[i], cout, B, cs[i].ic, cs[i].ih, cs[i].oc, cs[i].oh,
            900, cs[i].kh, cs[i].sh);
        cin = cout;
    }
    const float* c11 = cout;  // [B,128,1,900] in PING

    auto gemm = [&](const float* A, const float* Bm, float* C,
                    int M, int N, int K,
                    long Ars, long Acs, long Brs, long Bcs, long Crs, long Ccs,
                    long Ab, long Bb, long Cb, const float* bias, int relu, int nb) {
        dim3 g((N + 63) / 64, (M + 63) / 64, nb);
        if (Acs == 1 && Brs == 1)
            gemm_wmma_f32<true, true><<<g, 256, 0, stream>>>(A, Bm, C, M, N, K,
                Ars, Acs, Brs, Bcs, Crs, Ccs, Ab, Bb, Cb, bias, relu);
        else if (Acs == 1)
            gemm_wmma_f32<true, false><<<g, 256, 0, stream>>>(A, Bm, C, M, N, K,
                Ars, Acs, Brs, Bcs, Crs, Ccs, Ab, Bb, Cb, bias, relu);
        else if (Brs == 1)
            gemm_wmma_f32<false, true><<<g, 256, 0, stream>>>(A, Bm, C, M, N, K,
                Ars, Acs, Brs, Bcs, Crs, Ccs, Ab, Bb, Cb, bias, relu);
        else
            gemm_wmma_f32<false, false><<<g, 256, 0, stream>>>(A, Bm, C, M, N, K,
                Ars, Acs, Brs, Bcs, Crs, Ccs, Ab, Bb, Cb, bias, relu);
    };

    float* cat  = ws + OFF_CAT;   // [B,N,512]; [:,:256]=convLast1 out, [:,256:]=GAT out
    float* xh   = ws + OFF_XH;    // [B,N,1024]
    float* as_  = ws + OFF_AS;
    float* ad_  = ws + OFF_AD;
    float* outc = ws + OFF_OUTC;  // [B,N,1024]
    float* actb = ws + OFF_ACT;   // [B,N,64]
    float* asum = ws + OFF_ASUM;
    float* vlad = ws + OFF_VLAD;  // [B,65536]
    float* part = ws + OFF_PART;
    float* h1   = ws + OFF_H1;

    // convLast1: cat[b,n,o<256] = relu( sum_c c11[b,c,n] * W1[o,c] )
    gemm(c11, w_last1, cat, Nn, 256, 128,
         /*A*/ 1, 900, /*B*/ 1, 128, /*C*/ 512, 1,
         128 * 900, 0, (long)Nn * 512, nullptr, 1, B);

    // GAT projection: xh[b,n,hd] = sum_c cat[b,n,c] * gat_w[c,hd]
    gemm(cat, gat_w, xh, Nn, 1024, 256,
         512, 1, 1024, 1, 1024, 1,
         (long)Nn * 512, 0, (long)Nn * 1024, nullptr, 0, B);

    att_scores<<<B * Nn, 256, 0, stream>>>(xh, gat_asrc, gat_adst, as_, ad_);

    csr_count<<<(Nn + 255) / 256, 256, 0, stream>>>(ei, E, deg, Nn);
    csr_scan<<<1, 1, 0, stream>>>(deg, off, Nn);
    csr_fill<<<(Nn + 255) / 256, 256, 0, stream>>>(ei, E, off, csrc, Nn);

    {
        dim3 g(Nn, B);
        gat_aggregate<<<g, 256, 0, stream>>>(xh, as_, ad_, off, csrc, gat_bias, cat, Nn);
    }

    // convLast2: outc[b,n,o] = relu( sum_c cat[b,n,c] * W2[o,c] ), K=512, N=1024
    gemm(cat, w_last2, outc, Nn, 1024, 512,
         512, 1, 1, 512, 1024, 1,
         (long)Nn * 512, 0, (long)Nn * 1024, nullptr, 1, B);

    l2norm_rows<<<B * Nn, 256, 0, stream>>>(outc, 1024);

    // NetVLAD assignment: act = softmax( outc @ cluster_weights + cb )
    gemm(outc, cw, actb, Nn, 64, 1024,
         1024, 1, 64, 1, 64, 1,
         (long)Nn * 1024, 0, (long)Nn * 64, cb, 0, B);
    softmax64<<<B * Nn, 64, 0, stream>>>(actb);
    act_colsum<<<B, 64, 0, stream>>>(actb, asum, Nn);

    // vlad[b,f,k] = sum_n outc[b,n,f] * act[b,n,k]
    gemm(outc, actb, vlad, 1024, 64, Nn,
         1, 1024, 64, 1, 64, 1,
         (long)Nn * 1024, (long)Nn * 64, 65536, nullptr, 0, B);

    vlad_sub<<<(B * 65536) / 256, 256, 0, stream>>>(vlad, asum, cw2);
    l2norm_cols<<<B * 64, 256, 0, stream>>>(vlad);     // intra-normalization over f
    l2norm_rows<<<B, 256, 0, stream>>>(vlad, 65536);   // global per-batch normalize

    // hidden FC (memory-bound, deterministic split-K)
    {
        dim3 g(4, 8);
        hidden_partial<<<g, 256, 0, stream>>>(vlad, hw, part);
    }
    hidden_reduce<<<4, 256, 0, stream>>>(part, h1);

    bn_gate_final<<<1, 256, 0, stream>>>(h1, bn_g, bn_b, gate_w, gate_b, (float*)d_out);
}